// MultiAspectRetrieval_64390149702176
// MI455X (gfx1250) — compile-verified
//
#include <hip/hip_runtime.h>
#include <math.h>

// Problem constants (from reference)
#define S_ASP   4
#define DKD     64
#define DAD     1024
#define CCLUS   1024
#define NKEY    262144
#define MTOP    32
#define KMAXV   32
#define BATCH   32
#define NPC     (NKEY / CCLUS)     // 256
#define KCAT    (S_ASP * DKD)      // 256
#define EPSV    1e-8f

typedef __attribute__((ext_vector_type(2))) float v2f;
typedef __attribute__((ext_vector_type(8))) float v8f;

// ---------------------------------------------------------------------------
// CDNA5 async-copy helpers (GLOBAL_LOAD_ASYNC_TO_LDS_B128 + ASYNCcnt)
// ---------------------------------------------------------------------------
__device__ __forceinline__ unsigned lds_off(const void* p) {
  return (unsigned)(size_t)(const __attribute__((address_space(3))) void*)p;
}

__device__ __forceinline__ void async_ld_b128(unsigned loff, const float* g) {
  // copies 16 bytes from the global address g to the LDS address loff, tracked with ASYNCcnt
  asm volatile("global_load_async_to_lds_b128 %0, %1, off"
               :: "v"(loff), "v"(g)
               : "memory");
}

#if defined(__has_builtin)
#if __has_builtin(__builtin_amdgcn_s_wait_asynccnt)
#define WAIT_ASYNC(n) __builtin_amdgcn_s_wait_asynccnt(n)
#else
#define WAIT_ASYNC(n) asm volatile("s_wait_asynccnt %0" :: "i"(n) : "memory")
#endif
#else
#define WAIT_ASYNC(n) asm volatile("s_wait_asynccnt %0" :: "i"(n) : "memory")
#endif

// ---------------------------------------------------------------------------
// Kernel 1: queries_raw[sk][b] = sum_a W_Q[sk][a] * z[b][a]
// GEMM [256 x 1024] x [1024 x 32] via V_WMMA_F32_16X16X4_F32.
// ---------------------------------------------------------------------------
__global__ void qproj_kernel(const float* __restrict__ Wq,
                             const float* __restrict__ z,
                             float* __restrict__ qraw) {
  const int lane = threadIdx.x;
  const int half = lane >> 4;
  const int lrow = lane & 15;
  const int m0 = blockIdx.x * 16;   // row in [0,256)
  const int n0 = blockIdx.y * 16;   // col (batch) in [0,32)
  v8f acc = {};
  for (int kk = 0; kk < DAD; kk += 4) {
    const int ka = kk + 2 * half;
    v2f a, b;
    a.x = Wq[(m0 + lrow) * DAD + ka];
    a.y = Wq[(m0 + lrow) * DAD + ka + 1];
    b.x = z[(n0 + lrow) * DAD + ka];
    b.y = z[(n0 + lrow) * DAD + ka + 1];
    acc = __builtin_amdgcn_wmma_f32_16x16x4_f32(false, a, false, b,
                                                (short)0, acc, false, false);
  }
  for (int j = 0; j < 8; ++j) {
    const int m = m0 + j + 8 * half;
    qraw[m * BATCH + (n0 + lrow)] = acc[j];
  }
}

// ---------------------------------------------------------------------------
// Kernel 2: Qcat[b][s*64+d] = softmax(aw)[s] * q / (||q||+eps)
// ---------------------------------------------------------------------------
__global__ void qnorm_kernel(const float* __restrict__ qraw,
                             const float* __restrict__ aw,
                             float* __restrict__ Qcat) {
  const int bs = blockIdx.x;
  const int b = bs >> 2, s = bs & 3;
  const int d = threadIdx.x;
  const float v = qraw[(s * DKD + d) * BATCH + b];
  __shared__ float red[64];
  red[d] = v * v;
  __syncthreads();
  for (int off = 32; off > 0; off >>= 1) {
    if (d < off) red[d] += red[d + off];
    __syncthreads();
  }
  const float inv = 1.0f / (sqrtf(red[0]) + EPSV);
  const float a0 = aw[0], a1 = aw[1], a2 = aw[2], a3 = aw[3];
  const float mx = fmaxf(fmaxf(a0, a1), fmaxf(a2, a3));
  const float e0 = __expf(a0 - mx), e1 = __expf(a1 - mx),
              e2 = __expf(a2 - mx), e3 = __expf(a3 - mx);
  const float sum = e0 + e1 + e2 + e3;
  const float ws = ((s == 0) ? e0 : (s == 1) ? e1 : (s == 2) ? e2 : e3) / sum;
  Qcat[b * KCAT + s * DKD + d] = ws * v * inv;
}

// ---------------------------------------------------------------------------
// Kernel 3: Ccat[c][s*64+d] = centroids[s][c][d] / (||centroids[s][c]||+eps)
// ---------------------------------------------------------------------------
__global__ void cnorm_kernel(const float* __restrict__ cent,
                             float* __restrict__ Ccat) {
  const int cs = blockIdx.x;
  const int c = cs >> 2, s = cs & 3;
  const int d = threadIdx.x;
  const float v = cent[((long)s * CCLUS + c) * DKD + d];
  __shared__ float red[64];
  red[d] = v * v;
  __syncthreads();
  for (int off = 32; off > 0; off >>= 1) {
    if (d < off) red[d] += red[d + off];
    __syncthreads();
  }
  const float inv = 1.0f / (sqrtf(red[0]) + EPSV);
  Ccat[c * KCAT + s * DKD + d] = v * inv;
}

// ---------------------------------------------------------------------------
// Kernel 4: c_score[b][c] = Qcat[b] . Ccat[c]   (GEMM 32 x 1024, K=256)
// ---------------------------------------------------------------------------
__global__ void coarse_kernel(const float* __restrict__ Qcat,
                              const float* __restrict__ Ccat,
                              float* __restrict__ cscore) {
  const int lane = threadIdx.x;
  const int half = lane >> 4;
  const int lrow = lane & 15;
  const int m0 = blockIdx.y * 16;   // batch tile
  const int n0 = blockIdx.x * 16;   // centroid tile
  v8f acc = {};
  for (int kk = 0; kk < KCAT; kk += 4) {
    const int ka = kk + 2 * half;
    v2f a, b;
    a.x = Qcat[(m0 + lrow) * KCAT + ka];
    a.y = Qcat[(m0 + lrow) * KCAT + ka + 1];
    b.x = Ccat[(n0 + lrow) * KCAT + ka];
    b.y = Ccat[(n0 + lrow) * KCAT + ka + 1];
    acc = __builtin_amdgcn_wmma_f32_16x16x4_f32(false, a, false, b,
                                                (short)0, acc, false, false);
  }
  for (int j = 0; j < 8; ++j) {
    const int m = m0 + j + 8 * half;
    cscore[m * CCLUS + n0 + lrow] = acc[j];
  }
}

// ---------------------------------------------------------------------------
// Kernel 5: per-b top-32 clusters of c_score[b][0..1023]
// ---------------------------------------------------------------------------
__global__ void topk_clusters_kernel(const float* __restrict__ cscore,
                                     int* __restrict__ topclus) {
  const int b = blockIdx.x, t = threadIdx.x;
  float lv[4]; int li[4];
  for (int i = 0; i < 4; ++i) {
    const int c = t + i * 256;
    lv[i] = cscore[b * CCLUS + c];
    li[i] = c;
  }
  __shared__ float sv[256];
  __shared__ int   si[256];
  __shared__ int   winIdx;
  unsigned taken = 0;
  for (int k = 0; k < MTOP; ++k) {
    float bv = -INFINITY; int bi = 0x7fffffff;
    for (int i = 0; i < 4; ++i) {
      if (!((taken >> i) & 1u)) {
        if (lv[i] > bv || (lv[i] == bv && li[i] < bi)) { bv = lv[i]; bi = li[i]; }
      }
    }
    sv[t] = bv; si[t] = bi;
    __syncthreads();
    for (int off = 128; off > 0; off >>= 1) {
      if (t < off) {
        if (sv[t + off] > sv[t] || (sv[t + off] == sv[t] && si[t + off] < si[t])) {
          sv[t] = sv[t + off]; si[t] = si[t + off];
        }
      }
      __syncthreads();
    }
    if (t == 0) { topclus[b * MTOP + k] = si[0]; winIdx = si[0]; }
    __syncthreads();
    const int w = winIdx;
    for (int i = 0; i < 4; ++i)
      if (li[i] == w) taken |= (1u << i);
    __syncthreads();
  }
}

// ---------------------------------------------------------------------------
// Kernel 6: streaming fine scorer — the HBM-bound hot loop (256 MB pass).
// Persistent blocks, double-buffered LDS tiles filled with
// GLOBAL_LOAD_ASYNC_TO_LDS_B128 (ASYNCcnt), per-(key,aspect) inv-norm folded
// into the WMMA B operand. 2 waves x (M=16 batch rows), N=16 keys, K=256.
// ---------------------------------------------------------------------------
#define TKEYS        16
#define FINE_BLOCKS  2048
#define TILES        (NKEY / TKEYS)        // 16384
#define TPB          (TILES / FINE_BLOCKS) // 8
#define CHUNKS       ((TKEYS * KCAT) / 4)  // 1024 float4 chunks per tile

__device__ __forceinline__ void issue_tile(const float* __restrict__ pool,
                                           float (*kt)[KCAT], long n0, int t) {
  for (int idx = t; idx < CHUNKS; idx += 64) {
    const int s  = idx >> 8;            // 256 chunks per aspect
    const int j  = (idx >> 4) & (TKEYS - 1);
    const int c4 = idx & 15;
    const float* g = pool + ((long)s * NKEY + n0 + j) * DKD + c4 * 4;
    async_ld_b128(lds_off(&kt[j][s * DKD + c4 * 4]), g);
  }
}

__global__ void fine_kernel(const float* __restrict__ pool,
                            const float* __restrict__ Qcat,
                            float* __restrict__ s_all) {
  __shared__ float kt[2][TKEYS][KCAT];   // 2 x 16 KB double buffer
  __shared__ float rn[TKEYS][S_ASP];
  const int t = threadIdx.x;             // 0..63 (2 waves)
  const int tile0 = blockIdx.x * TPB;

  issue_tile(pool, kt[0], (long)tile0 * TKEYS, t);

  for (int it = 0; it < TPB; ++it) {
    const int tile = tile0 + it;
    const int buf = it & 1;
    if (it + 1 < TPB) {
      issue_tile(pool, kt[buf ^ 1], (long)(tile + 1) * TKEYS, t);
      WAIT_ASYNC(16);  // 16 outstanding = next tile only -> current complete
    } else {
      WAIT_ASYNC(0);
    }
    __syncthreads();

    // Per-(key, aspect) inverse norms: 64 pairs, 64 threads.
    {
      const int j = t >> 2, s = t & 3;
      float a = 0.f;
      for (int d = 0; d < DKD; ++d) {
        const float x = kt[buf][j][s * DKD + d];
        a += x * x;
      }
      rn[j][s] = 1.0f / (sqrtf(a) + EPSV);
    }
    __syncthreads();

    // WMMA: wave w owns batch tile m0 = w*16; 16 keys in N.
    const int w    = t >> 5;
    const int lane = t & 31;
    const int half = lane >> 4;
    const int lrow = lane & 15;
    const int m0   = w * 16;
    float r4[S_ASP];
    for (int s = 0; s < S_ASP; ++s) r4[s] = rn[lrow][s];
    v8f acc = {};
    const float* qrow = Qcat + (m0 + lrow) * KCAT;
    for (int s = 0; s < S_ASP; ++s) {
      const float r = r4[s];
      for (int kk = s * DKD; kk < (s + 1) * DKD; kk += 4) {
        const int ka = kk + 2 * half;
        v2f a, b;
        a.x = qrow[ka];
        a.y = qrow[ka + 1];
        b.x = kt[buf][lrow][ka] * r;
        b.y = kt[buf][lrow][ka + 1] * r;
        acc = __builtin_amdgcn_wmma_f32_16x16x4_f32(false, a, false, b,
                                                    (short)0, acc, false, false);
      }
    }
    const long n0 = (long)tile * TKEYS;
    for (int j2 = 0; j2 < 8; ++j2) {
      const int brow = m0 + j2 + 8 * half;
      s_all[(long)brow * NKEY + n0 + lrow] = acc[j2];
    }
    __syncthreads();   // all waves done reading buf before it is refilled
  }
}

// ---------------------------------------------------------------------------
// Kernel 7: per-b finalize (gather candidates, gate, normalize, top-32).
// Output: [alphas 32*32][global_idx 32*32 (as float)][soft_full 32*8192]
// ---------------------------------------------------------------------------
__global__ void finalize_kernel(const float* __restrict__ s_all,
                                const int* __restrict__ topclus,
                                const float* __restrict__ tau_p,
                                const int* __restrict__ lam_p,
                                const int* __restrict__ warm_p,
                                float* __restrict__ out) {
  const int b = blockIdx.x, t = threadIdx.x;
  const float tau = *tau_p;
  const float lam = (float)(*lam_p);
  const int warm = *warm_p;
  __shared__ int clus[32];
  if (t < 32) clus[t] = topclus[b * MTOP + t];
  __syncthreads();

  float ss[32];
  float sv[32];
  float local_sum = 0.f, local_max = -INFINITY;
  for (int i = 0; i < 32; ++i) {
    const int p = i * 256 + t;
    const int ci = p >> 8, j = p & 255;
    const int n = clus[ci] * NPC + j;
    const float s = s_all[(long)b * NKEY + n];
    ss[i] = s;
    if (warm) {
      local_max = fmaxf(local_max, s);
    } else {
      const float g = 1.0f / (1.0f + __expf(-lam * (s - tau)));
      const float r = g * __expf(s);    // T = 1
      sv[i] = r;
      local_sum += r;
    }
  }

  __shared__ float red[256];
  float smax = 0.f;
  if (warm) {
    red[t] = local_max;
    __syncthreads();
    for (int off = 128; off > 0; off >>= 1) {
      if (t < off) red[t] = fmaxf(red[t], red[t + off]);
      __syncthreads();
    }
    smax = red[0];
    __syncthreads();
    local_sum = 0.f;
    for (int i = 0; i < 32; ++i) local_sum += __expf(ss[i] - smax);
  }
  red[t] = local_sum;
  __syncthreads();
  for (int off = 128; off > 0; off >>= 1) {
    if (t < off) red[t] += red[t + off];
    __syncthreads();
  }
  const float S1 = red[0];
  __syncthreads();

  float sel[32];
  for (int i = 0; i < 32; ++i) {
    const int p = i * 256 + t;
    float soft;
    if (warm) soft = __expf(ss[i] - smax) / S1;
    else      soft = sv[i] / (S1 + EPSV);
    out[2 * BATCH * MTOP + (long)b * (MTOP * NPC) + p] = soft;
    sel[i] = warm ? ss[i] : soft;
  }

  __shared__ float rv[256];
  __shared__ int   ri[256];
  __shared__ float wv[32];
  __shared__ int   wp[32];
  unsigned taken = 0;
  for (int k = 0; k < KMAXV; ++k) {
    float bv = -INFINITY; int bp = 0x7fffffff;
    for (int i = 0; i < 32; ++i) {
      if (!((taken >> i) & 1u)) {
        const float v = sel[i];
        const int p = i * 256 + t;
        if (v > bv || (v == bv && p < bp)) { bv = v; bp = p; }
      }
    }
    rv[t] = bv; ri[t] = bp;
    __syncthreads();
    for (int off = 128; off > 0; off >>= 1) {
      if (t < off) {
        if (rv[t + off] > rv[t] || (rv[t + off] == rv[t] && ri[t + off] < ri[t])) {
          rv[t] = rv[t + off]; ri[t] = ri[t + off];
        }
      }
      __syncthreads();
    }
    if (t == 0) { wv[k] = rv[0]; wp[k] = ri[0]; }
    __syncthreads();
    const int pw = wp[k];
    if ((pw & 255) == t) taken |= (1u << (pw >> 8));
    __syncthreads();
  }

  if (t < KMAXV) {
    const float v = wv[t];
    const int p = wp[t];
    const int n = clus[p >> 8] * NPC + (p & 255);
    float alpha;
    if (warm) {
      float mx = -INFINITY;
      for (int k = 0; k < KMAXV; ++k) mx = fmaxf(mx, wv[k]);
      float s2 = 0.f;
      for (int k = 0; k < KMAXV; ++k) s2 += __expf(wv[k] - mx);
      alpha = __expf(v - mx) / s2;
    } else {
      float s2 = 0.f;
      for (int k = 0; k < KMAXV; ++k) s2 += wv[k];
      alpha = v / (s2 + EPSV);
    }
    out[b * MTOP + t] = alpha;
    out[BATCH * MTOP + b * MTOP + t] = (float)n;
  }
}

// ---------------------------------------------------------------------------
extern "C" void kernel_launch(void* const* d_in, const int* in_sizes, int n_in,
                              void* d_out, int out_size, void* d_ws, size_t ws_size,
                              hipStream_t stream) {
  (void)in_sizes; (void)n_in; (void)out_size; (void)ws_size;
  const float* z    = (const float*)d_in[0];   // [32, 1024]
  const float* pool = (const float*)d_in[1];   // [4, 262144, 64]
  const float* Wq   = (const float*)d_in[2];   // [4, 64, 1024]
  const float* aw   = (const float*)d_in[3];   // [4]
  const float* tau  = (const float*)d_in[4];   // scalar
  const float* cent = (const float*)d_in[5];   // [4, 1024, 64]
  const int* lam    = (const int*)d_in[6];     // scalar int
  const int* warm   = (const int*)d_in[7];     // scalar int

  float* ws = (float*)d_ws;
  float* qraw    = ws;                  //   8192 f
  float* Qcat    = ws + 8192;           //   8192 f
  float* Ccat    = ws + 16384;          // 262144 f
  float* cscore  = ws + 278528;         //  32768 f
  int*   topclus = (int*)(ws + 311296); //   1024 i
  float* s_all   = ws + 312320;         // 32*262144 f (32 MB)
  float* out = (float*)d_out;

  qproj_kernel<<<dim3(16, 2), 32, 0, stream>>>(Wq, z, qraw);
  qnorm_kernel<<<BATCH * S_ASP, 64, 0, stream>>>(qraw, aw, Qcat);
  cnorm_kernel<<<CCLUS * S_ASP, 64, 0, stream>>>(cent, Ccat);
  coarse_kernel<<<dim3(64, 2), 32, 0, stream>>>(Qcat, Ccat, cscore);
  topk_clusters_kernel<<<BATCH, 256, 0, stream>>>(cscore, topclus);
  fine_kernel<<<FINE_BLOCKS, 64, 0, stream>>>(pool, Qcat, s_all);
  finalize_kernel<<<BATCH, 256, 0, stream>>>(s_all, topclus, tau, lam, warm, out);
}